// NimbleLoss_17772574671032
// MI455X (gfx1250) — compile-verified
//
#include <hip/hip_runtime.h>
#include <math.h>

#define CANVAS_H 28
#define CANVAS_W 28
#define NPIX (CANVAS_H * CANVAS_W)   // 784
#define NPTS 128
#define NSEG (NPTS - 1)              // 127
#define EPS_F 1e-7f

typedef __attribute__((ext_vector_type(2))) float v2f;
typedef __attribute__((ext_vector_type(8))) float v8f;

__device__ __forceinline__ int iabs_i(int v) { return v < 0 ? -v : v; }

// One workgroup per image: rasterize into an LDS canvas, fuse coord-MSE and
// binary-BCE, emit per-image partial sums.
__global__ __launch_bounds__(256) void raster_loss_kernel(
    const float* __restrict__ pred,   // [B,128,2]
    const float* __restrict__ targ,   // [B,128,2]
    const float* __restrict__ tbmp,   // [B,28,28]
    float* __restrict__ coordP,       // [B] partial sum of squared diffs
    float* __restrict__ bceP)         // [B] partial sum of bce
{
    const int b = blockIdx.x;
    const int t = threadIdx.x;

    __shared__ int sx[NPTS];
    __shared__ int sy[NPTS];
    __shared__ unsigned char canvas[NPIX];
    __shared__ float redC[8];
    __shared__ float redB[8];

    // Zero canvas (dword-wide).
    {
        unsigned int* c32 = (unsigned int*)canvas;
        for (int i = t; i < NPIX / 4; i += 256) c32[i] = 0u;
    }

    // Load pred coords, quantize to pixel grid (truncation, per reference).
    if (t < NPTS) {
        float px = pred[(b * NPTS + t) * 2 + 0];
        float py = pred[(b * NPTS + t) * 2 + 1];
        sx[t] = (int)(px * (float)(CANVAS_W - 1));
        sy[t] = (int)(py * (float)(CANVAS_H - 1));
    }

    // Prefetch this block's target bitmap (3136 B) while we rasterize.
    {
        const float* tb = tbmp + (size_t)b * NPIX;
        if (t < 49) __builtin_prefetch(tb + t * 16, 0, 1);  // 64B per line
    }

    __syncthreads();

    // Rasterize 127 segments x up-to-28 steps (MAXI = 28).
    for (int j = t; j < NSEG * CANVAS_W; j += 256) {
        int seg = j / CANVAS_W;
        int i   = j - seg * CANVAS_W;
        int x0 = sx[seg],     y0 = sy[seg];
        int x1 = sx[seg + 1], y1 = sy[seg + 1];
        bool steep = iabs_i(y1 - y0) > iabs_i(x1 - x0);
        int X0 = steep ? y0 : x0;
        int Y0 = steep ? x0 : y0;
        int X1 = steep ? y1 : x1;
        int Y1 = steep ? x1 : y1;
        if (X0 > X1) { int s = X0; X0 = X1; X1 = s; s = Y0; Y0 = Y1; Y1 = s; }
        int dx = X1 - X0;
        int dy = iabs_i(Y1 - Y0);
        int ystep = (Y0 < Y1) ? 1 : -1;
        int dxs = dx > 1 ? dx : 1;                 // max(dx,1)
        int x = X0 + i;
        int y = Y0 + (int)((float)(ystep * dy * i) / (float)dxs);  // trunc
        int px = steep ? y : x;
        int py = steep ? x : y;
        if (i <= dx && px >= 0 && px < CANVAS_W && py >= 0 && py < CANVAS_H)
            canvas[py * CANVAS_W + px] = 1;        // all writers store 1: race-free
    }

    // Coord MSE partial: 256 components per image, one per thread.
    float cpart;
    {
        float d = pred[(size_t)b * 256 + t] - targ[(size_t)b * 256 + t];
        cpart = d * d;
    }

    __syncthreads();

    // Binary BCE: clip makes p in {EPS, 1-EPS}; logs are compile-time constants.
    const float logE   = logf(EPS_F);
    const float log1mE = logf(1.0f - EPS_F);
    float bpart = 0.0f;
    for (int p = t; p < NPIX; p += 256) {
        float tb  = tbmp[(size_t)b * NPIX + p];
        bool hit  = canvas[p] != 0;
        float lp  = hit ? log1mE : logE;   // log(p)
        float l1p = hit ? logE   : log1mE; // log(1-p)
        bpart -= tb * lp + (1.0f - tb) * l1p;
    }

    // Block reduction: wave32 shuffles, then 8 wave sums via LDS.
    for (int off = 16; off > 0; off >>= 1) {
        cpart += __shfl_down(cpart, off, 32);
        bpart += __shfl_down(bpart, off, 32);
    }
    int wave = t >> 5;
    int lane = t & 31;
    if (lane == 0) { redC[wave] = cpart; redB[wave] = bpart; }
    __syncthreads();
    if (t == 0) {
        float c = 0.0f, bb = 0.0f;
#pragma unroll
        for (int w = 0; w < 8; w++) { c += redC[w]; bb += redB[w]; }
        coordP[b] = c;
        bceP[b]   = bb;
    }
}

// Single-wave finalize: fold B partials with V_WMMA_F32_16X16X4_F32 chains.
// B = all-ones => D[m][n] = rowsum(A[m]) + C[m][n] for every column n, so the
// A-operand element placement is irrelevant for the total, and
// sum(all 256 C entries) == 16 * grand_total.
__global__ __launch_bounds__(32) void finalize_kernel(
    const float* __restrict__ coordP,
    const float* __restrict__ bceP,
    float* __restrict__ out,
    int B)
{
    const int lane = threadIdx.x;   // exactly one wave32, EXEC all-ones
    v2f ones; ones.x = 1.0f; ones.y = 1.0f;
    v8f cc = {};
    v8f cb = {};
    for (int base = 0; base < B; base += 64) {
        v2f a, bv;
        a.x  = coordP[base + lane];
        a.y  = coordP[base + 32 + lane];
        bv.x = bceP[base + lane];
        bv.y = bceP[base + 32 + lane];
        cc = __builtin_amdgcn_wmma_f32_16x16x4_f32(false, a,  false, ones,
                                                   (short)0, cc, false, false);
        cb = __builtin_amdgcn_wmma_f32_16x16x4_f32(false, bv, false, ones,
                                                   (short)0, cb, false, false);
    }
    float sc = 0.0f, sb = 0.0f;
#pragma unroll
    for (int r = 0; r < 8; r++) { sc += cc[r]; sb += cb[r]; }
    for (int off = 16; off > 0; off >>= 1) {
        sc += __shfl_down(sc, off, 32);
        sb += __shfl_down(sb, off, 32);
    }
    if (lane == 0) {
        float coord_loss  = (sc * (1.0f / 16.0f)) / ((float)B * (float)(NPTS * 2));
        float raster_loss = (sb * (1.0f / 16.0f)) / ((float)B * (float)NPIX);
        out[0] = coord_loss;
        out[1] = raster_loss;
        out[2] = 1.0f * coord_loss + 0.5f * raster_loss;
    }
}

extern "C" void kernel_launch(void* const* d_in, const int* in_sizes, int n_in,
                              void* d_out, int out_size, void* d_ws, size_t ws_size,
                              hipStream_t stream) {
    const float* pred = (const float*)d_in[0];   // [B,128,2]
    const float* targ = (const float*)d_in[1];   // [B,128,2]
    const float* tbmp = (const float*)d_in[2];   // [B,28,28]
    float* outp = (float*)d_out;

    const int B = in_sizes[0] / (NPTS * 2);      // 4096

    float* coordP = (float*)d_ws;                // [B]
    float* bceP   = coordP + B;                  // [B]

    raster_loss_kernel<<<B, 256, 0, stream>>>(pred, targ, tbmp, coordP, bceP);
    finalize_kernel<<<1, 32, 0, stream>>>(coordP, bceP, outp, B);
}